// PAA_79929341378891
// MI455X (gfx1250) — compile-verified
//
#include <hip/hip_runtime.h>
#include <hip/hip_bf16.h>
#include <math.h>

#define HWSZ   65536            // H*W
#define CDIM   64
#define BDIM   2
#define NTOT   (BDIM * HWSZ)    // 131072 pixels
#define CHW    (CDIM * HWSZ)    // per-batch plane stride

typedef __attribute__((ext_vector_type(2))) float v2f;
typedef __attribute__((ext_vector_type(8))) float v8f;

// ---------------------------------------------------------------------------
// K1: GeM pooling: a[b,c] = (mean_hw clip(x,1e-6)^p[c]) ^ (1/p[c])
// ---------------------------------------------------------------------------
__global__ void gem_kernel(const float* __restrict__ x,
                           const float* __restrict__ gem_p,
                           float* __restrict__ a) {
  int bc = blockIdx.x;                 // 0..127 == b*64+c
  int c  = bc & 63;
  const float* xp = x + (long)bc * HWSZ;
  float pexp = gem_p[c];
  float sum = 0.f;
  for (int i = threadIdx.x; i < HWSZ; i += 256) {
    float v = fmaxf(xp[i], 1e-6f);
    sum += __powf(v, pexp);
  }
  __shared__ float red[256];
  red[threadIdx.x] = sum;
  __syncthreads();
  for (int s2 = 128; s2 > 0; s2 >>= 1) {
    if (threadIdx.x < s2) red[threadIdx.x] += red[threadIdx.x + s2];
    __syncthreads();
  }
  if (threadIdx.x == 0)
    a[bc] = __powf(red[0] * (1.0f / HWSZ), 1.0f / pexp);
}

// ---------------------------------------------------------------------------
// K2: gate MLP: sigmoid(w2 @ relu(bn(w1 @ a + b1)) + b2)   (128 values total)
// ---------------------------------------------------------------------------
__global__ void gate_kernel(const float* __restrict__ a,
                            const float* __restrict__ w1, const float* __restrict__ b1,
                            const float* __restrict__ g,  const float* __restrict__ bb,
                            const float* __restrict__ m,  const float* __restrict__ vv,
                            const float* __restrict__ w2, const float* __restrict__ b2,
                            float* __restrict__ gate) {
  __shared__ float sa[128], sh[128];
  int t = threadIdx.x;                 // blockDim = 128
  sa[t] = a[t];
  __syncthreads();
  {
    int b = t >> 6, o = t & 63;
    float acc = b1[o];
    for (int c = 0; c < 64; ++c) acc += w1[o * 64 + c] * sa[b * 64 + c];
    float sc = g[o] * rsqrtf(vv[o] + 1e-5f);
    acc = (acc - m[o]) * sc + bb[o];
    sh[t] = fmaxf(acc, 0.f);
  }
  __syncthreads();
  {
    int b = t >> 6, o = t & 63;
    float acc = b2[o];
    for (int c = 0; c < 64; ++c) acc += w2[o * 64 + c] * sh[b * 64 + c];
    gate[t] = 1.f / (1.f + __expf(-acc));
  }
}

// ---------------------------------------------------------------------------
// K3: elementwise: out = x*(1+gate[b,c]);  xb = BN(x)
// ---------------------------------------------------------------------------
__global__ void ew_kernel(const float* __restrict__ x,
                          const float* __restrict__ bn_g, const float* __restrict__ bn_b,
                          const float* __restrict__ bn_m, const float* __restrict__ bn_v,
                          const float* __restrict__ gate,
                          float* __restrict__ out, float* __restrict__ xb) {
  long i = (long)blockIdx.x * 256 + threadIdx.x;
  int c = (int)((i >> 16) & 63);
  int b = (int)(i >> 22);
  float xv = x[i];
  float sc = bn_g[c] * rsqrtf(bn_v[c] + 1e-5f);
  out[i] = xv * (1.f + gate[b * 64 + c]);
  xb[i]  = (xv - bn_m[c]) * sc + bn_b[c];
}

// ---------------------------------------------------------------------------
// K4: q-projection GEMM  P[b,o,hw] = sum_c W[o,c] * X[b,c,hw]
// One wave: 16-pixel column tile x all 64 output rows (4 WMMA row-tiles).
// fp32 WMMA 16x16x4; weights staged in LDS (A-frags via ds_load_b64).
// ---------------------------------------------------------------------------
__global__ void __launch_bounds__(256) qproj_kernel(const float* __restrict__ W,
                                                    const float* __restrict__ X,
                                                    float* __restrict__ P) {
  __shared__ float sW[64 * 64];
  for (int i = threadIdx.x; i < 4096; i += 256) sW[i] = W[i];
  __syncthreads();
  int wave = threadIdx.x >> 5, lane = threadIdx.x & 31;
  int half = lane >> 4, lcol = lane & 15;
  long n0 = ((long)blockIdx.x * 8 + wave) * 16;
  int b   = (int)(n0 >> 16);
  int pix = (int)(n0 & (HWSZ - 1)) + lcol;
  const float* xb = X + (long)b * CHW + pix;
  v8f acc0 = {}, acc1 = {}, acc2 = {}, acc3 = {};
#pragma unroll 4
  for (int k0 = 0; k0 < 64; k0 += 4) {
    int kb = k0 + 2 * half;                       // lanes 0-15: K=k0,k0+1 ; 16-31: K=k0+2,k0+3
    v2f bf;
    bf.x = xb[(long)kb * HWSZ];
    bf.y = xb[(long)(kb + 1) * HWSZ];
    if (k0 < 60) __builtin_prefetch(&xb[(long)(kb + 4) * HWSZ], 0, 0);
    v2f a0 = *(const v2f*)&sW[( 0 + lcol) * 64 + kb];
    v2f a1 = *(const v2f*)&sW[(16 + lcol) * 64 + kb];
    v2f a2 = *(const v2f*)&sW[(32 + lcol) * 64 + kb];
    v2f a3 = *(const v2f*)&sW[(48 + lcol) * 64 + kb];
    acc0 = __builtin_amdgcn_wmma_f32_16x16x4_f32(false, a0, false, bf, (short)0, acc0, false, false);
    acc1 = __builtin_amdgcn_wmma_f32_16x16x4_f32(false, a1, false, bf, (short)0, acc1, false, false);
    acc2 = __builtin_amdgcn_wmma_f32_16x16x4_f32(false, a2, false, bf, (short)0, acc2, false, false);
    acc3 = __builtin_amdgcn_wmma_f32_16x16x4_f32(false, a3, false, bf, (short)0, acc3, false, false);
  }
  float* pout = P + (long)b * CHW + pix;
#pragma unroll
  for (int v = 0; v < 8; ++v) {
    int mrow = v + 8 * half;                      // D: M = vgpr + 8*laneHalf
    pout[(long)(mrow)      * HWSZ] = acc0[v];
    pout[(long)(mrow + 16) * HWSZ] = acc1[v];
    pout[(long)(mrow + 32) * HWSZ] = acc2[v];
    pout[(long)(mrow + 48) * HWSZ] = acc3[v];
  }
}

// ---------------------------------------------------------------------------
// K5: attention scores  F[b,n,hw] = sum_c P[b,c,hw] * P[b,c,tap_n(hw)]
// ---------------------------------------------------------------------------
__global__ void score_kernel(const float* __restrict__ P, float* __restrict__ F, int r) {
  int idx = blockIdx.x * 256 + threadIdx.x;       // 0..131071
  int b = idx >> 16;
  int hw = idx & (HWSZ - 1);
  int h = hw >> 8, w = hw & 255;
  const float* pb = P + (long)b * CHW + hw;
  int off[9]; bool val[9];
  int n = 0;
#pragma unroll
  for (int di = -1; di <= 1; ++di)
#pragma unroll
    for (int dj = -1; dj <= 1; ++dj) {
      int hh = h + di * r, ww = w + dj * r;
      val[n] = (hh >= 0 && hh < 256 && ww >= 0 && ww < 256);
      off[n] = di * r * 256 + dj * r;
      ++n;
    }
  float acc[9] = {0, 0, 0, 0, 0, 0, 0, 0, 0};
  for (int c = 0; c < 64; ++c) {
    float pc = pb[c << 16];
#pragma unroll
    for (int t = 0; t < 9; ++t) {
      float tv = val[t] ? pb[(c << 16) + off[t]] : 0.f;
      acc[t] += pc * tv;
    }
  }
#pragma unroll
  for (int t = 0; t < 9; ++t)
    F[((long)(b * 9 + t) << 16) + hw] = acc[t];
}

// ---------------------------------------------------------------------------
// K6: depthwise ks x ks conv per tap + 9x9 mix + softmax over taps
// ---------------------------------------------------------------------------
__global__ void pe_softmax_kernel(const float* __restrict__ F, const float* __restrict__ DW,
                                  const float* __restrict__ PW, float* __restrict__ S,
                                  int ks, int pad) {
  __shared__ float sdw[9 * 5 * 5];
  __shared__ float spw[81];
  int nk = 9 * ks * ks;
  for (int i = threadIdx.x; i < nk; i += 256) sdw[i] = DW[i];
  for (int i = threadIdx.x; i < 81; i += 256) spw[i] = PW[i];
  __syncthreads();
  int idx = blockIdx.x * 256 + threadIdx.x;
  int b = idx >> 16;
  int hw = idx & (HWSZ - 1);
  int h = hw >> 8, w = hw & 255;
  float y[9];
#pragma unroll
  for (int t = 0; t < 9; ++t) {
    float acc = 0.f;
    const float* fb = F + ((long)(b * 9 + t) << 16);
    for (int u = 0; u < ks; ++u) {
      int hh = h + u - pad;
      if (hh < 0 || hh > 255) continue;
      for (int v = 0; v < ks; ++v) {
        int ww = w + v - pad;
        if (ww < 0 || ww > 255) continue;
        acc += fb[hh * 256 + ww] * sdw[t * ks * ks + u * ks + v];
      }
    }
    y[t] = acc;
  }
  float z[9];
  float mx = -1e30f;
#pragma unroll
  for (int o = 0; o < 9; ++o) {
    float acc = 0.f;
#pragma unroll
    for (int t = 0; t < 9; ++t) acc += spw[o * 9 + t] * y[t];
    z[o] = acc;
    mx = fmaxf(mx, acc);
  }
  float sum = 0.f;
#pragma unroll
  for (int o = 0; o < 9; ++o) { z[o] = __expf(z[o] - mx); sum += z[o]; }
  float inv = 1.f / sum;
#pragma unroll
  for (int o = 0; o < 9; ++o)
    S[((long)(b * 9 + o) << 16) + hw] = z[o] * inv;
}

// ---------------------------------------------------------------------------
// K7: fused weighted-tap-sum + output-projection WMMA GEMM, accumulated into OUT
// B-fragment o[c,pix] = sum_n P[b,c,tap_n] * S[b,n,pix] built on the fly.
// ---------------------------------------------------------------------------
__global__ void __launch_bounds__(256) applyproj_kernel(const float* __restrict__ P,
                                                        const float* __restrict__ S,
                                                        const float* __restrict__ W,
                                                        const float* __restrict__ BIAS,
                                                        float* __restrict__ OUT, int r) {
  __shared__ float sW[64 * 64];
  __shared__ float sB[64];
  for (int i = threadIdx.x; i < 4096; i += 256) sW[i] = W[i];
  if (threadIdx.x < 64) sB[threadIdx.x] = BIAS[threadIdx.x];
  __syncthreads();
  int wave = threadIdx.x >> 5, lane = threadIdx.x & 31;
  int half = lane >> 4, lcol = lane & 15;
  long n0 = ((long)blockIdx.x * 8 + wave) * 16;
  int b   = (int)(n0 >> 16);
  int pix = (int)(n0 & (HWSZ - 1)) + lcol;
  int h = pix >> 8, w = pix & 255;

  float sv[9]; bool val[9]; int off[9];
  const float* sbase = S + ((long)(b * 9) << 16) + pix;
  int n = 0;
#pragma unroll
  for (int di = -1; di <= 1; ++di)
#pragma unroll
    for (int dj = -1; dj <= 1; ++dj) {
      int hh = h + di * r, ww = w + dj * r;
      val[n] = (hh >= 0 && hh < 256 && ww >= 0 && ww < 256);
      off[n] = di * r * 256 + dj * r;
      sv[n]  = sbase[(long)n << 16];
      ++n;
    }
  const float* pbase = P + (long)b * CHW + pix;
  v8f acc0 = {}, acc1 = {}, acc2 = {}, acc3 = {};
  for (int k0 = 0; k0 < 64; k0 += 4) {
    int kb = k0 + 2 * half;
    const float* pc = pbase + ((long)kb << 16);
    float o0 = 0.f, o1 = 0.f;
#pragma unroll
    for (int t = 0; t < 9; ++t) {
      if (val[t]) {
        o0 += pc[off[t]]        * sv[t];
        o1 += pc[off[t] + HWSZ] * sv[t];
      }
    }
    v2f bf; bf.x = o0; bf.y = o1;
    v2f a0 = *(const v2f*)&sW[( 0 + lcol) * 64 + kb];
    v2f a1 = *(const v2f*)&sW[(16 + lcol) * 64 + kb];
    v2f a2 = *(const v2f*)&sW[(32 + lcol) * 64 + kb];
    v2f a3 = *(const v2f*)&sW[(48 + lcol) * 64 + kb];
    acc0 = __builtin_amdgcn_wmma_f32_16x16x4_f32(false, a0, false, bf, (short)0, acc0, false, false);
    acc1 = __builtin_amdgcn_wmma_f32_16x16x4_f32(false, a1, false, bf, (short)0, acc1, false, false);
    acc2 = __builtin_amdgcn_wmma_f32_16x16x4_f32(false, a2, false, bf, (short)0, acc2, false, false);
    acc3 = __builtin_amdgcn_wmma_f32_16x16x4_f32(false, a3, false, bf, (short)0, acc3, false, false);
  }
  float* outb = OUT + (long)b * CHW + pix;
#pragma unroll
  for (int v = 0; v < 8; ++v) {
    int m0 = v + 8 * half;
    { int oc = m0;      long o = (long)oc << 16; outb[o] += acc0[v] + sB[oc]; }
    { int oc = m0 + 16; long o = (long)oc << 16; outb[o] += acc1[v] + sB[oc]; }
    { int oc = m0 + 32; long o = (long)oc << 16; outb[o] += acc2[v] + sB[oc]; }
    { int oc = m0 + 48; long o = (long)oc << 16; outb[o] += acc3[v] + sB[oc]; }
  }
}

// ---------------------------------------------------------------------------
extern "C" void kernel_launch(void* const* d_in, const int* in_sizes, int n_in,
                              void* d_out, int out_size, void* d_ws, size_t ws_size,
                              hipStream_t stream) {
  const float* x     = (const float*)d_in[0];
  const float* bn_g  = (const float*)d_in[1];
  const float* bn_b  = (const float*)d_in[2];
  const float* bn_m  = (const float*)d_in[3];
  const float* bn_v  = (const float*)d_in[4];
  const float* gem_p = (const float*)d_in[5];
  const float* ga_w1 = (const float*)d_in[6];
  const float* ga_b1 = (const float*)d_in[7];
  const float* ga_bg = (const float*)d_in[8];
  const float* ga_bb = (const float*)d_in[9];
  const float* ga_bm = (const float*)d_in[10];
  const float* ga_bv = (const float*)d_in[11];
  const float* ga_w2 = (const float*)d_in[12];
  const float* ga_b2 = (const float*)d_in[13];

  float* ws   = (float*)d_ws;
  float* xb   = ws;                    // 8388608 floats
  float* p    = xb + 8388608;          // 8388608
  float* f    = p + 8388608;           // 1179648
  float* s    = f + 1179648;           // 1179648
  float* a    = s + 1179648;           // 128
  float* gate = a + 128;               // 128

  float* out = (float*)d_out;

  gem_kernel<<<128, 256, 0, stream>>>(x, gem_p, a);
  gate_kernel<<<1, 128, 0, stream>>>(a, ga_w1, ga_b1, ga_bg, ga_bb, ga_bm, ga_bv,
                                     ga_w2, ga_b2, gate);
  ew_kernel<<<32768, 256, 0, stream>>>(x, bn_g, bn_b, bn_m, bn_v, gate, out, xb);

  const int rates[3] = {1, 3, 5};
  for (int i = 0; i < 3; ++i) {
    const float* qw = (const float*)d_in[14 + 5 * i];
    const float* dw = (const float*)d_in[15 + 5 * i];
    const float* pw = (const float*)d_in[16 + 5 * i];
    const float* ow = (const float*)d_in[17 + 5 * i];
    const float* ob = (const float*)d_in[18 + 5 * i];
    int r  = rates[i];
    int ks = 2 * (r / 2) + 1;
    int pad = r / 2;
    qproj_kernel<<<1024, 256, 0, stream>>>(qw, xb, p);
    score_kernel<<<512, 256, 0, stream>>>(p, f, r);
    pe_softmax_kernel<<<512, 256, 0, stream>>>(f, dw, pw, s, ks, pad);
    applyproj_kernel<<<1024, 256, 0, stream>>>(p, s, ow, ob, out, r);
  }
}